// MoE_CNN_94489281288
// MI455X (gfx1250) — compile-verified
//
#include <hip/hip_runtime.h>

// ---------------------------------------------------------------------------
// MoE-CNN forward for MI455X (gfx1250).
//   conv1d(CH=128,K=32,stride=8), 320 out-channels (64 gate + 256 expert),
//   B=64, L=2048, L'=253, then max(relu(.)) over positions + tiny MLP tail.
// Conv mapped to V_WMMA_F32_16X16X4_F32 (exact f32):
//   - weights pre-packed fragment-major  -> coalesced A loads
//   - input window staged in LDS via global_load_async_to_lds_b128 (ASYNCcnt)
//   - 5 waves/WG, 4 m-tiles/wave sharing one B fragment from LDS
// ---------------------------------------------------------------------------

typedef float v2f __attribute__((ext_vector_type(2)));
typedef float v8f __attribute__((ext_vector_type(8)));

#define BB    64
#define LL    2048
#define CHN   128
#define KK    32
#define STR   8
#define LOUT  253            // (2048-32)/8 + 1
#define NT    16             // n-tiles of 16 positions (last tile clamped)
#define MT    20             // m-tiles: 320 out-channels / 16
#define KRED  4096           // 128*32 reduction depth
#define OCH   320
#define NSTEP 1024           // KRED / 4
#define SXPAD 156            // LDS row stride (152 samples + pad)
#define WTILE 65536          // floats per packed m-tile: 1024 steps * 64

// ---------------------------------------------------------------------------
// Kernel 1: pack fs/fp [B, L, 64] -> X [B, 128, 2048]  (LDS-tiled transpose)
// ---------------------------------------------------------------------------
__global__ void pack_features(const float* __restrict__ fs,
                              const float* __restrict__ fp,
                              float* __restrict__ X) {
    __shared__ float tile[32][33];
    const int ltile = blockIdx.x, ctile = blockIdx.y, b = blockIdx.z;
    const int tx = threadIdx.x, ty = threadIdx.y;

    const int l = ltile * 32 + ty;
    const int c = ctile * 32 + tx;
    const float* src = (c < 64) ? fs : fp;
    tile[ty][tx] = src[((size_t)b * LL + l) * 64 + (c & 63)];
    __syncthreads();

    const int c2 = ctile * 32 + ty;
    const int l2 = ltile * 32 + tx;
    X[((size_t)b * CHN + c2) * LL + l2] = tile[tx][ty];
}

// ---------------------------------------------------------------------------
// Kernel 2: pack conv weights into WMMA-A fragment-major order.
//   Wpk[((mtile*1024 + s)*32 + lane)*2 + j] =
//       Wstacked[mtile*16 + (lane&15)][s*4 + (lane>>4)*2 + j]
//   -> per (mtile, k-step) the 32-lane b64 A-load is 256B contiguous.
// ---------------------------------------------------------------------------
__global__ void pack_weights(const float* __restrict__ gate_cw,  // [64,4096]
                             const float* __restrict__ exp_cw,   // [256,4096]
                             float* __restrict__ Wpk) {
    const int idx  = blockIdx.x * 256 + threadIdx.x;   // < 20*1024*64
    const int j    = idx & 1;
    const int lane = (idx >> 1) & 31;
    const int s    = (idx >> 6) & 1023;
    const int mt   = idx >> 16;
    const int kidx = s * 4 + (lane >> 4) * 2 + j;
    const int row  = mt * 16 + (lane & 15);
    Wpk[idx] = (mt < 4) ? gate_cw[(size_t)row * KRED + kidx]
                        : exp_cw[(size_t)(row - 64) * KRED + kidx];
}

// ---------------------------------------------------------------------------
// Async stage of one input window (128 ch x NV4*4 samples) into LDS.
// Uses CDNA5 GLOBAL_LOAD_ASYNC_TO_LDS_B128: VDST = LDS byte offset (VGPR),
// VADDR = 64-bit global address, no SADDR.  Tracked by ASYNCcnt.
// ---------------------------------------------------------------------------
template <int NV4>
__device__ __forceinline__ void stage_async(const float* __restrict__ g,
                                            float* sX, int tid) {
    for (int u = tid; u < CHN * NV4; u += 160) {
        const int c = u / NV4;
        const int i = (u - c * NV4) * 4;
        const unsigned lds_off = (unsigned)(size_t)(sX + c * SXPAD + i);
        const float* gp = g + (size_t)c * LL + i;
        asm volatile("global_load_async_to_lds_b128 %0, %1, off"
                     :: "v"(lds_off), "v"(gp) : "memory");
    }
    asm volatile("s_wait_asynccnt 0" ::: "memory");
}

__device__ __forceinline__ void reduce_store(v8f acc, int col, int half,
                                             float* __restrict__ dst) {
    // max over the 16 columns of the C/D tile; xor masks 1,2,4,8 stay inside
    // each 16-lane half (row m = v + 8*half, col = lane&15).
#pragma unroll
    for (int v = 0; v < 8; ++v) {
        float m = acc[v];
        m = fmaxf(m, __shfl_xor(m, 1));
        m = fmaxf(m, __shfl_xor(m, 2));
        m = fmaxf(m, __shfl_xor(m, 4));
        m = fmaxf(m, __shfl_xor(m, 8));
        if (col == 0) dst[half * 8 + v] = m;
    }
}

// ---------------------------------------------------------------------------
// Kernel 3: WMMA conv + per-tile max.  Grid (NT, BB), block = 160 (5 waves).
//   Wave w computes m-tiles 4w..4w+3; all waves share the LDS input window.
// ---------------------------------------------------------------------------
__global__ __launch_bounds__(160)
void conv_wmma(const float* __restrict__ X,     // [B,128,2048]
               const float* __restrict__ Wpk,   // [20][1024][64]
               float* __restrict__ partial) {   // [B, NT, 320]
    __shared__ float sX[CHN * SXPAD];           // 79,872 B

    const int ntile = blockIdx.x;
    const int b     = blockIdx.y;
    const int tid   = threadIdx.x;
    const int lane  = tid & 31;
    const int wave  = tid >> 5;                 // 0..4 -> m-group

    const float* g = X + (size_t)b * CHN * LL + ntile * 128;  // t0*8 samples
    if (ntile != NT - 1) stage_async<38>(g, sX, tid);         // 152 samples
    else                 stage_async<32>(g, sX, tid);         // 128 valid
    __syncthreads();

    const int col = lane & 15, half = lane >> 4, half2 = half * 2;
    int tcl = ntile * 16 + col;
    if (tcl > LOUT - 1) tcl = LOUT - 1;         // clamp: duplicate cols are
    const int bbase = (tcl - ntile * 16) * STR + half2;  // max-neutral

    const float* wa = Wpk + (size_t)(wave * 4) * WTILE + lane * 2;

    v8f a0 = {0,0,0,0,0,0,0,0}, a1 = a0, a2 = a0, a3 = a0;

    for (int c = 0; c < CHN; ++c) {
        const float* sxc = sX + c * SXPAD + bbase;
        const float* wac = wa + c * 512;        // 8 steps * 64 floats
#pragma unroll
        for (int k4 = 0; k4 < 8; ++k4) {
            v2f bv = *(const v2f*)(sxc + k4 * 4);           // ds_load_b64
            v2f f0 = *(const v2f*)(wac + 0 * WTILE + k4 * 64);
            v2f f1 = *(const v2f*)(wac + 1 * WTILE + k4 * 64);
            v2f f2 = *(const v2f*)(wac + 2 * WTILE + k4 * 64);
            v2f f3 = *(const v2f*)(wac + 3 * WTILE + k4 * 64);
            a0 = __builtin_amdgcn_wmma_f32_16x16x4_f32(false, f0, false, bv, (short)0, a0, false, false);
            a1 = __builtin_amdgcn_wmma_f32_16x16x4_f32(false, f1, false, bv, (short)0, a1, false, false);
            a2 = __builtin_amdgcn_wmma_f32_16x16x4_f32(false, f2, false, bv, (short)0, a2, false, false);
            a3 = __builtin_amdgcn_wmma_f32_16x16x4_f32(false, f3, false, bv, (short)0, a3, false, false);
        }
    }

    float* base = partial + ((size_t)b * NT + ntile) * OCH + wave * 64;
    reduce_store(a0, col, half, base + 0);
    reduce_store(a1, col, half, base + 16);
    reduce_store(a2, col, half, base + 32);
    reduce_store(a3, col, half, base + 48);
}

// ---------------------------------------------------------------------------
// Kernel 4: tail — max over n-tiles + bias + relu, gate softmax, expert MLPs,
// mixture, fc + log_softmax.  One block per batch row, 320 threads.
// ---------------------------------------------------------------------------
__global__ void moe_tail(const float* __restrict__ partial,  // [B,NT,320]
                         const float* __restrict__ gate_cb,
                         const float* __restrict__ gate_lw,
                         const float* __restrict__ gate_lb,
                         const float* __restrict__ exp_cb,
                         const float* __restrict__ exp_w1,
                         const float* __restrict__ exp_b1,
                         const float* __restrict__ exp_w2,
                         const float* __restrict__ exp_b2,
                         const float* __restrict__ fc_w,
                         const float* __restrict__ fc_b,
                         float* __restrict__ out) {           // [640 | 256]
    __shared__ float s_conv[OCH];
    __shared__ float s_glog[4], s_gate[4];
    __shared__ float s_h1[128], s_h2[64], s_mix[16], s_fc[10], s_lse;

    const int b = blockIdx.x;
    const int tid = threadIdx.x;

    if (tid < OCH) {
        const float* p = partial + (size_t)b * NT * OCH + tid;
        float m = -__builtin_inff();
        for (int nt = 0; nt < NT; ++nt) m = fmaxf(m, p[nt * OCH]);
        float bias = (tid < 64) ? gate_cb[tid] : exp_cb[tid - 64];
        s_conv[tid] = fmaxf(m + bias, 0.f);
    }
    __syncthreads();

    if (tid < 4) {
        float a = gate_lb[tid];
        const float* w = gate_lw + tid * 64;
        for (int c = 0; c < 64; ++c) a += w[c] * s_conv[c];
        s_glog[tid] = a;
    }
    __syncthreads();
    if (tid == 0) {
        float mx = s_glog[0];
        for (int e = 1; e < 4; ++e) mx = fmaxf(mx, s_glog[e]);
        float s = 0.f, ex[4];
        for (int e = 0; e < 4; ++e) { ex[e] = __expf(s_glog[e] - mx); s += ex[e]; }
        for (int e = 0; e < 4; ++e) {
            s_gate[e] = ex[e] / s;
            out[640 + b * 4 + e] = s_gate[e];
        }
    }
    __syncthreads();

    if (tid < 128) {
        const int e = tid >> 5, o = tid & 31;
        float a = exp_b1[e * 32 + o];
        const float* w = exp_w1 + (size_t)(e * 32 + o) * 64;
        const float* h = s_conv + 64 + e * 64;
        for (int c = 0; c < 64; ++c) a += w[c] * h[c];
        s_h1[tid] = fmaxf(a, 0.f);
    }
    __syncthreads();

    if (tid < 64) {
        const int e = tid >> 4, pq = tid & 15;
        float a = exp_b2[e * 16 + pq];
        const float* w = exp_w2 + (size_t)(e * 16 + pq) * 32;
        const float* h = s_h1 + e * 32;
        for (int o = 0; o < 32; ++o) a += w[o] * h[o];
        s_h2[tid] = fmaxf(a, 0.f);
    }
    __syncthreads();

    if (tid < 16) {
        float a = 0.f;
        for (int e = 0; e < 4; ++e) a += s_gate[e] * s_h2[e * 16 + tid];
        s_mix[tid] = a;
    }
    __syncthreads();

    if (tid < 10) {
        float a = fc_b[tid];
        const float* w = fc_w + tid * 16;
        for (int f = 0; f < 16; ++f) a += w[f] * s_mix[f];
        s_fc[tid] = a;
    }
    __syncthreads();
    if (tid == 0) {
        float mx = s_fc[0];
        for (int o = 1; o < 10; ++o) mx = fmaxf(mx, s_fc[o]);
        float s = 0.f;
        for (int o = 0; o < 10; ++o) s += __expf(s_fc[o] - mx);
        s_lse = mx + __logf(s);
    }
    __syncthreads();
    if (tid < 10) out[b * 10 + tid] = s_fc[tid] - s_lse;
}

// ---------------------------------------------------------------------------
extern "C" void kernel_launch(void* const* d_in, const int* in_sizes, int n_in,
                              void* d_out, int out_size, void* d_ws, size_t ws_size,
                              hipStream_t stream) {
    const float* fs      = (const float*)d_in[0];
    const float* fp      = (const float*)d_in[1];
    const float* gate_cw = (const float*)d_in[2];
    const float* gate_cb = (const float*)d_in[3];
    const float* gate_lw = (const float*)d_in[4];
    const float* gate_lb = (const float*)d_in[5];
    const float* exp_cw  = (const float*)d_in[6];
    const float* exp_cb  = (const float*)d_in[7];
    const float* exp_w1  = (const float*)d_in[8];
    const float* exp_b1  = (const float*)d_in[9];
    const float* exp_w2  = (const float*)d_in[10];
    const float* exp_b2  = (const float*)d_in[11];
    const float* fc_w    = (const float*)d_in[12];
    const float* fc_b    = (const float*)d_in[13];
    float* out = (float*)d_out;

    // scratch: X [16.8M floats] | Wpk [1.31M floats] | partial [0.33M floats]
    float* Xp      = (float*)d_ws;
    float* Wpk     = Xp + (size_t)BB * CHN * LL;
    float* partial = Wpk + (size_t)MT * NSTEP * 64;

    pack_features<<<dim3(LL / 32, CHN / 32, BB), dim3(32, 32), 0, stream>>>(fs, fp, Xp);
    pack_weights<<<(MT * NSTEP * 64) / 256, 256, 0, stream>>>(gate_cw, exp_cw, Wpk);
    conv_wmma<<<dim3(NT, BB), 160, 0, stream>>>(Xp, Wpk, partial);
    moe_tail<<<BB, OCH, 0, stream>>>(partial, gate_cb, gate_lw, gate_lb, exp_cb,
                                     exp_w1, exp_b1, exp_w2, exp_b2, fc_w, fc_b, out);
}